// GraphNet_11441792877370
// MI455X (gfx1250) — compile-verified
//
#include <hip/hip_runtime.h>
#include <hip/hip_bf16.h>
#include <math.h>

typedef float v2f __attribute__((ext_vector_type(2)));
typedef float v8f __attribute__((ext_vector_type(8)));

#define EPS_BN 1e-5f
#define LDS_STRIDE 136   // 128 + 8 pad: half-wave row pairs hit disjoint LDS banks

// ---------------------------------------------------------------------------
// init: deg=1 (self loop), zero stat + pool accumulators
// ---------------------------------------------------------------------------
__global__ void k_init(float* __restrict__ deg, float* __restrict__ sum,
                       float* __restrict__ sumsq, float* __restrict__ psums,
                       float* __restrict__ pcnt, int N, int G, int D) {
    int i = blockIdx.x * blockDim.x + threadIdx.x;
    if (i < N) deg[i] = 1.0f;                 // self-loop contributes 1 to degree
    if (i < G * D) psums[i] = 0.0f;
    if (i < G) pcnt[i] = 0.0f;
    if (i < D) { sum[i] = 0.0f; sumsq[i] = 0.0f; }
}

// ---------------------------------------------------------------------------
// xw = x @ W via V_WMMA_F32_16X16X4_F32.
// One wave owns a 16x128 strip (8 accumulators); W staged in LDS via the
// gfx1250 async-to-LDS path (ASYNCcnt), so each A fragment is loaded once.
// ---------------------------------------------------------------------------
__global__ void k_gemm_xw(const float* __restrict__ X, const float* __restrict__ W,
                          float* __restrict__ XW, int Nrows, int D) {
    __shared__ float wlds[128 * LDS_STRIDE];

    // cooperative async copy of W (128x128 f32) into LDS, padded stride.
    // 4096 float4 chunks over 256 threads -> 16 async b128 loads per thread.
    {
        int t = threadIdx.x;
        #pragma unroll
        for (int i = 0; i < 16; ++i) {
            int c  = t + i * 256;             // float4 chunk id
            int r  = c >> 5;                  // row (32 chunks per row)
            int cw = (c & 31) << 2;           // starting column
            unsigned lofs = (unsigned)(size_t)&wlds[r * LDS_STRIDE + cw];
            const float* g = W + (size_t)r * 128 + cw;
            asm volatile("global_load_async_to_lds_b128 %0, %1, off"
                         :: "v"(lofs), "v"(g) : "memory");
        }
        asm volatile("s_wait_asynccnt 0" ::: "memory");
    }
    __syncthreads();

    int strip = blockIdx.x * 8 + (threadIdx.x >> 5);
    int lane  = threadIdx.x & 31;
    int row0  = strip << 4;
    if (row0 >= Nrows) return;                // wave-uniform: EXEC all-1s at WMMA
    int m    = lane & 15;
    int half = lane >> 4;
    int klo  = half << 1;

    const float* arow = X + (size_t)(row0 + m) * D;
    v8f acc[8] = {};
    for (int k = 0; k < 128; k += 4) {
        v2f a;
        a.x = arow[k + klo];                  // one b64 load, reused for 8 tiles
        a.y = arow[k + klo + 1];
        #pragma unroll
        for (int nt = 0; nt < 8; ++nt) {
            v2f b;
            b.x = wlds[(k + klo) * LDS_STRIDE + (nt << 4) + m];
            b.y = wlds[(k + klo + 1) * LDS_STRIDE + (nt << 4) + m];
            acc[nt] = __builtin_amdgcn_wmma_f32_16x16x4_f32(false, a, false, b,
                                                            (short)0, acc[nt],
                                                            false, false);
        }
    }
    #pragma unroll
    for (int nt = 0; nt < 8; ++nt) {
        #pragma unroll
        for (int r = 0; r < 8; ++r) {
            int mm = r + (half << 3);
            XW[(size_t)(row0 + mm) * D + (nt << 4) + m] = acc[nt][r];
        }
    }
}

// ---------------------------------------------------------------------------
// degree count over edge destinations
// ---------------------------------------------------------------------------
__global__ void k_deg(const int* __restrict__ dst, float* __restrict__ deg, int E) {
    int e = blockIdx.x * blockDim.x + threadIdx.x;
    if (e < E) atomicAdd(&deg[dst[e]], 1.0f);
}

__global__ void k_dinv(float* __restrict__ deg, int N) {
    int i = blockIdx.x * blockDim.x + threadIdx.x;
    if (i < N) {
        float d = deg[i];
        deg[i] = d > 0.0f ? rsqrtf(d) : 0.0f;   // in place: deg -> dinv
    }
}

// out[v] = xw[v] * dinv[v]^2   (self-loop term; initializes 'out' fully)
__global__ void k_selfloop(const float* __restrict__ xw, const float* __restrict__ dinv,
                           float* __restrict__ out, int N, int D) {
    int gid = blockIdx.x * blockDim.x + threadIdx.x;
    int v = gid >> 7;            // D == 128
    int d = gid & 127;
    if (v < N) {
        float di = dinv[v];
        out[(size_t)v * D + d] = xw[(size_t)v * D + d] * di * di;
    }
}

// one wave per edge: lane loads float4 (b128) of xw[src], atomic-adds to out[dst].
// xw (51MB) and out (51MB) both fit in MI455X's 192MB L2 -> on-chip traffic.
__global__ void k_edge_scatter(const int* __restrict__ src, const int* __restrict__ dst,
                               const float* __restrict__ xw, const float* __restrict__ dinv,
                               float* __restrict__ out, int E, int D) {
    int gid  = blockIdx.x * blockDim.x + threadIdx.x;
    int e    = gid >> 5;
    int lane = threadIdx.x & 31;
    if (e >= E) return;
    if (lane == 0) {                          // stream-ahead on the index arrays
        __builtin_prefetch(&src[e + 4096], 0, 0);   // speculative: OOB dropped
        __builtin_prefetch(&dst[e + 4096], 0, 0);
    }
    int s = src[e];
    int t = dst[e];
    float nrm = dinv[s] * dinv[t];
    const float4* xs = (const float4*)(xw + (size_t)s * D);
    float4 v = xs[lane];                      // 32 lanes * 4 = 128 floats
    float* o = out + (size_t)t * D + lane * 4;
    atomicAdd(o + 0, v.x * nrm);
    atomicAdd(o + 1, v.y * nrm);
    atomicAdd(o + 2, v.z * nrm);
    atomicAdd(o + 3, v.w * nrm);
}

// ---------------------------------------------------------------------------
// BN stats: per-channel sum / sumsq (block partials in LDS -> 128-word atomics).
// Note: the '+ b' bias cancels exactly inside BN (shifts mean only), so skipped.
// ---------------------------------------------------------------------------
__global__ void k_bn_stats(const float* __restrict__ out, float* __restrict__ sum,
                           float* __restrict__ sumsq, int N, int D, int rowsPerBlock) {
    __shared__ float ls[256];
    __shared__ float lq[256];
    int t  = threadIdx.x;
    int d  = t & 127;
    int rh = t >> 7;                         // 0 or 1
    int row0 = blockIdx.x * rowsPerBlock;
    int rend = min(row0 + rowsPerBlock, N);
    float s = 0.0f, q = 0.0f;
    for (int r = row0 + rh; r < rend; r += 2) {
        float v = out[(size_t)r * D + d];
        s += v; q += v * v;
    }
    ls[t] = s; lq[t] = q;
    __syncthreads();
    if (t < 128) {
        atomicAdd(&sum[d],   ls[t] + ls[t + 128]);
        atomicAdd(&sumsq[d], lq[t] + lq[t + 128]);
    }
}

// scale/shift so that normalized = out*scale + shift  (gamma/beta folded in)
__global__ void k_bn_finalize(const float* __restrict__ sum, const float* __restrict__ sumsq,
                              const float* __restrict__ gamma, const float* __restrict__ beta,
                              float* __restrict__ scale, float* __restrict__ shift, int N, int D) {
    int d = blockIdx.x * blockDim.x + threadIdx.x;
    if (d < D) {
        float invN = 1.0f / (float)N;
        float mean = sum[d] * invN;
        float var  = sumsq[d] * invN - mean * mean;
        float sc   = gamma[d] * rsqrtf(var + EPS_BN);
        scale[d] = sc;
        shift[d] = beta[d] - mean * sc;
    }
}

// normalize + ELU + mean-pool accumulation (batch is sorted; 32KB pool region
// stays hot in L2 for the f32 atomics)
__global__ void k_norm_elu_pool(const float* __restrict__ out, const float* __restrict__ scale,
                                const float* __restrict__ shift, const int* __restrict__ batch,
                                float* __restrict__ psums, float* __restrict__ pcnt,
                                int N, int D) {
    int gid = blockIdx.x * blockDim.x + threadIdx.x;
    int v = gid >> 7;
    int d = gid & 127;
    if (v >= N) return;
    float y = out[(size_t)v * D + d] * scale[d] + shift[d];
    y = y > 0.0f ? y : (__expf(y) - 1.0f);    // ELU, alpha=1 (v_exp_f32)
    int g = batch[v];
    atomicAdd(&psums[(size_t)g * D + d], y);
    if (d == 0) atomicAdd(&pcnt[g], 1.0f);
}

// ---------------------------------------------------------------------------
// result[g][j] = (psums[g][:]/max(cnt,1)) . fc_W[j][:] + fc_b[j]  via WMMA
// ---------------------------------------------------------------------------
__global__ void k_final_gemm(const float* __restrict__ psums, const float* __restrict__ pcnt,
                             const float* __restrict__ fcW, const float* __restrict__ fcb,
                             float* __restrict__ OUT, int G, int D) {
    int wave = (blockIdx.x * blockDim.x + threadIdx.x) >> 5;
    int lane = threadIdx.x & 31;
    int tilesN = D >> 4;
    int mt = wave / tilesN;
    int nt = wave - mt * tilesN;
    int row0 = mt << 4;
    if (row0 >= G) return;                    // wave-uniform
    int col0 = nt << 4;
    int m    = lane & 15;
    int half = lane >> 4;
    int klo  = half << 1;

    float inv = 1.0f / fmaxf(pcnt[row0 + m], 1.0f);   // fold mean-pool division
    const float* arow = psums + (size_t)(row0 + m) * D;
    v8f acc = {};
    for (int k = 0; k < D; k += 4) {
        v2f a, b;
        a.x = arow[k + klo] * inv;
        a.y = arow[k + klo + 1] * inv;
        b.x = fcW[(size_t)(col0 + m) * D + k + klo];      // B = fc_W^T
        b.y = fcW[(size_t)(col0 + m) * D + k + klo + 1];
        acc = __builtin_amdgcn_wmma_f32_16x16x4_f32(false, a, false, b,
                                                    (short)0, acc, false, false);
    }
    #pragma unroll
    for (int r = 0; r < 8; ++r) {
        int mm = r + (half << 3);
        OUT[(size_t)(row0 + mm) * D + col0 + m] = acc[r] + fcb[col0 + m];
    }
}

// ---------------------------------------------------------------------------
extern "C" void kernel_launch(void* const* d_in, const int* in_sizes, int n_in,
                              void* d_out, int out_size, void* d_ws, size_t ws_size,
                              hipStream_t stream) {
    const float* x     = (const float*)d_in[0];
    const int*   eidx  = (const int*)d_in[1];
    const int*   batch = (const int*)d_in[2];
    const float* W     = (const float*)d_in[3];
    // d_in[4] = b : cancels exactly inside BatchNorm (mean shift), unused
    const float* gamma = (const float*)d_in[5];
    const float* beta  = (const float*)d_in[6];
    const float* fcW   = (const float*)d_in[7];
    const float* fcb   = (const float*)d_in[8];
    float* outp = (float*)d_out;

    const int N = in_sizes[2];          // 100000
    const int D = in_sizes[4];          // 128
    const int E = in_sizes[1] / 2;      // 1.6M
    const int G = out_size / D;         // 64

    const int* src = eidx;
    const int* dst = eidx + E;

    // workspace layout (floats)
    float* ws    = (float*)d_ws;
    float* xw    = ws;                            // N*D
    float* out   = xw + (size_t)N * D;            // N*D
    float* deg   = out + (size_t)N * D;           // N   (becomes dinv in place)
    float* sum   = deg + N;                       // D
    float* sumsq = sum + D;                       // D
    float* scale = sumsq + D;                     // D
    float* shift = scale + D;                     // D
    float* psums = shift + D;                     // G*D
    float* pcnt  = psums + (size_t)G * D;         // G

    // 1. init accumulators / degree
    k_init<<<(N + 255) / 256, 256, 0, stream>>>(deg, sum, sumsq, psums, pcnt, N, G, D);

    // 2. xw = x @ W   (one 16x128 strip per wave, W async-staged in LDS)
    {
        int strips = N / 16;
        int blocks = (strips + 7) / 8;            // 8 waves / block
        k_gemm_xw<<<blocks, 256, 0, stream>>>(x, W, xw, N, D);
    }

    // 3. degree -> dinv
    k_deg<<<(E + 255) / 256, 256, 0, stream>>>(dst, deg, E);
    k_dinv<<<(N + 255) / 256, 256, 0, stream>>>(deg, N);

    // 4. self-loop init + edge gather/scatter
    k_selfloop<<<((size_t)N * D / 256 + 1), 256, 0, stream>>>(xw, deg, out, N, D);
    {
        long long thr = (long long)E * 32;
        int blocks = (int)((thr + 255) / 256);
        k_edge_scatter<<<blocks, 256, 0, stream>>>(src, dst, xw, deg, out, E, D);
    }

    // 5. BatchNorm stats + scale/shift
    {
        const int rpb = 512;
        int blocks = (N + rpb - 1) / rpb;
        k_bn_stats<<<blocks, 256, 0, stream>>>(out, sum, sumsq, N, D, rpb);
        k_bn_finalize<<<1, 128, 0, stream>>>(sum, sumsq, gamma, beta, scale, shift, N, D);
    }

    // 6. normalize + ELU + pool accumulation
    {
        long long thr = (long long)N * D;
        int blocks = (int)((thr + 255) / 256);
        k_norm_elu_pool<<<blocks, 256, 0, stream>>>(out, scale, shift, batch, psums, pcnt, N, D);
    }

    // 7. pooled @ fc_W^T + fc_b
    {
        int tiles = (G / 16) * (D / 16);          // 32 waves
        int blocks = (tiles + 7) / 8;
        k_final_gemm<<<blocks, 256, 0, stream>>>(psums, pcnt, fcW, fcb, outp, G, D);
    }
    (void)n_in; (void)ws_size;
}